// top_k_pool_18013092839704
// MI455X (gfx1250) — compile-verified
//
#include <hip/hip_runtime.h>
#include <stdint.h>

// Problem constants from the reference: B=16, N=2048, D=256, K=512
namespace {
constexpr int BB = 16;
constexpr int NN = 2048;
constexpr int DD = 256;
constexpr int KK = 512;
}

typedef unsigned long long u64_t;

typedef unsigned int u32x4 __attribute__((ext_vector_type(4)));
typedef int          i32x8 __attribute__((ext_vector_type(8)));
typedef int          i32x4 __attribute__((ext_vector_type(4)));
typedef int          v4i   __attribute__((vector_size(16)));   // gcc-style v4i32

#if defined(__HIP_DEVICE_COMPILE__) && \
    __has_builtin(__builtin_amdgcn_tensor_load_to_lds) && \
    __has_builtin(__builtin_amdgcn_tensor_store_from_lds) && \
    __has_builtin(__builtin_amdgcn_s_wait_tensorcnt)
#define TOPK_USE_TDM 1
#else
#define TOPK_USE_TDM 0
#endif

#if defined(__HIP_DEVICE_COMPILE__) && \
    __has_builtin(__builtin_amdgcn_global_load_async_to_lds_b128) && \
    __has_builtin(__builtin_amdgcn_s_wait_asynccnt)
#define TOPK_USE_ASYNC_LDS 1
#else
#define TOPK_USE_ASYNC_LDS 0
#endif

// Map float -> u32 such that unsigned ascending order == score DESCENDING order.
// Combined with the index in the low 32 bits, an ascending u64 sort reproduces
// jax.lax.top_k ordering (desc value, asc index on ties) exactly.
__device__ __forceinline__ unsigned order_key_desc(float f) {
  unsigned x = __float_as_uint(f);
  unsigned y = (x & 0x80000000u) ? ~x : (x | 0x80000000u); // ascending float order
  return ~y;                                               // flip -> descending
}

// ---------------------------------------------------------------------------
// Kernel 1: per-batch bitonic full sort of 2048 (key,idx) pairs in LDS.
// 1024 threads = 32 wave32 waves per workgroup; 16 KB LDS.
// Emits ss (top-K values, sorted) and the selected indices to workspace.
// ---------------------------------------------------------------------------
__global__ __launch_bounds__(1024) void topk_kernel(
    const float* __restrict__ scores, int* __restrict__ idx_ws,
    float* __restrict__ ss_out) {
  __shared__ u64_t keys[NN];
  const int b = blockIdx.x;
  const int t = threadIdx.x;
  const float* s = scores + (size_t)b * NN;

  for (int i = t; i < NN; i += 1024)
    keys[i] = ((u64_t)order_key_desc(s[i]) << 32) | (unsigned)i;
  __syncthreads();

  for (int k = 2; k <= NN; k <<= 1) {
    for (int j = k >> 1; j > 0; j >>= 1) {
      unsigned i = ((t & ~(unsigned)(j - 1)) << 1) | (t & (unsigned)(j - 1));
      unsigned l = i | (unsigned)j;
      u64_t a = keys[i], c = keys[l];
      bool up = ((i & (unsigned)k) == 0);
      if ((a > c) == up) { keys[i] = c; keys[l] = a; }
      __syncthreads();
    }
  }

  if (t < KK) {
    unsigned sel = (unsigned)(keys[t] & 0xFFFFFFFFu);
    idx_ws[b * KK + t] = (int)sel;
    ss_out[b * KK + t] = s[sel];  // exact value, no decode round-trip
  }
}

// ---------------------------------------------------------------------------
// Kernel 2: hs gather. One wave per 16 selected rows.
// gfx1250 path: TDM gather-mode load (16 row indices per D#) into LDS, then
// TDM 1-D tile store to the contiguous output region. s_wait_tensorcnt syncs.
// ---------------------------------------------------------------------------
__global__ __launch_bounds__(32) void gather_h_kernel(
    const float* __restrict__ h, const int* __restrict__ idx_ws,
    float* __restrict__ hs_out) {
  __shared__ float tile[16 * DD];  // 16 KB staging
  const int blk = blockIdx.x;
  const int b   = blk / (KK / 16);
  const int grp = blk % (KK / 16);
  const int* idx = idx_ws + b * KK + grp * 16;  // uniform address -> scalar loads

#if TOPK_USE_TDM
  const unsigned lds_off = (unsigned)(uintptr_t)&tile[0];  // LDS aperture: addr[31:0]
  const uint64_t gaddr = (uint64_t)(uintptr_t)(h + (size_t)b * NN * DD);

  // ---- D# for gather-mode load: tensor = 2048 x 256 f32, tile = 16 rows x 256
  u32x4 g0;
  g0[0] = 0x80000001u;  // count=1, gather_mode=1 (bit31), 16-bit indices (bit30=0)
  g0[1] = lds_off;
  g0[2] = (unsigned)(gaddr & 0xFFFFFFFFu);
  g0[3] = (unsigned)((gaddr >> 32) & 0x01FFFFFFu) | (2u << 30);  // type=2 "image"

  i32x8 g1;
  g1[0] = (int)(2u << 16);                       // data_size=2 (4 bytes), mask=0
  g1[1] = (int)(((unsigned)DD & 0xFFFFu) << 16); // tensor_dim0 lo16 = 256
  g1[2] = (int)(((unsigned)NN & 0xFFFFu) << 16); // dim0 hi=0 | tensor_dim1 lo16 = 2048
  g1[3] = (int)(((unsigned)DD & 0xFFFFu) << 16); // dim1 hi=0 | tile_dim0 = 256
  g1[4] = 16;                                    // tile_dim1 = #valid indices
  g1[5] = DD;                                    // tensor_dim0_stride = 256
  g1[6] = 0;
  g1[7] = 0;

  i32x4 g2, g3;                                  // 16 packed 16-bit row indices
  g2[0] = idx[0]  | (idx[1]  << 16);
  g2[1] = idx[2]  | (idx[3]  << 16);
  g2[2] = idx[4]  | (idx[5]  << 16);
  g2[3] = idx[6]  | (idx[7]  << 16);
  g3[0] = idx[8]  | (idx[9]  << 16);
  g3[1] = idx[10] | (idx[11] << 16);
  g3[2] = idx[12] | (idx[13] << 16);
  g3[3] = idx[14] | (idx[15] << 16);

  i32x8 z8;
  z8[0] = 0; z8[1] = 0; z8[2] = 0; z8[3] = 0;
  z8[4] = 0; z8[5] = 0; z8[6] = 0; z8[7] = 0;

  __builtin_amdgcn_tensor_load_to_lds(g0, g1, g2, g3, z8, 0);
  __builtin_amdgcn_s_wait_tensorcnt(0);  // LDS tile ready

  // ---- D# for store: contiguous 1-D tile of 16*256 = 4096 f32 to output
  const uint64_t oaddr =
      (uint64_t)(uintptr_t)(hs_out + ((size_t)b * KK + (size_t)grp * 16) * DD);
  constexpr unsigned TOT = 16u * DD;  // 4096
  u32x4 s0;
  s0[0] = 1u;  // count=1, normal mode
  s0[1] = lds_off;
  s0[2] = (unsigned)(oaddr & 0xFFFFFFFFu);
  s0[3] = (unsigned)((oaddr >> 32) & 0x01FFFFFFu) | (2u << 30);

  i32x8 s1;
  s1[0] = (int)(2u << 16);               // data_size = 4 bytes
  s1[1] = (int)((TOT & 0xFFFFu) << 16);  // tensor_dim0 = 4096
  s1[2] = (int)(1u << 16);               // tensor_dim1 = 1
  s1[3] = (int)((TOT & 0xFFFFu) << 16);  // tile_dim0 = 4096
  s1[4] = 0;                             // tile_dim1 unused (1-D tile)
  s1[5] = (int)TOT;                      // tensor_dim0_stride
  s1[6] = 0;
  s1[7] = 0;

  i32x4 z4; z4[0] = 0; z4[1] = 0; z4[2] = 0; z4[3] = 0;
  __builtin_amdgcn_tensor_store_from_lds(s0, s1, z4, z4, z8, 0);
  __builtin_amdgcn_s_wait_tensorcnt(0);
#else
  // Fallback: direct float4 row copies (same semantics, plain VMEM).
  const int lane = threadIdx.x;
  for (int r = 0; r < 16; ++r) {
    const float4* src =
        (const float4*)(h + ((size_t)b * NN + (size_t)idx[r]) * DD);
    float4* dst =
        (float4*)(hs_out + ((size_t)b * KK + (size_t)grp * 16 + r) * DD);
    for (int c = lane; c < DD / 4; c += 32) dst[c] = src[c];
  }
  (void)tile;
#endif
}

// ---------------------------------------------------------------------------
// Kernel 3: gs gather: out[b,i,j] = g[b, idx[i], idx[j]].
// Stage the selected 8 KB row + the 512 indices in LDS; scattered accesses
// become LDS reads, global reads/writes stay fully coalesced.
// gfx1250 path: row staged via GLOBAL_LOAD_ASYNC_TO_LDS_B128 (ASYNCcnt),
// bypassing VGPRs entirely.
// ---------------------------------------------------------------------------
__global__ __launch_bounds__(256) void gather_g_kernel(
    const float* __restrict__ g, const int* __restrict__ idx_ws,
    float* __restrict__ gs_out) {
  __shared__ int   idxs[KK];  // 2 KB
  __shared__ float row[NN];   // 8 KB
  const int b = blockIdx.y;
  const int i = blockIdx.x;
  const int t = threadIdx.x;
  const int* idx = idx_ws + b * KK;

  for (int j = t; j < KK; j += 256) idxs[j] = idx[j];
  __syncthreads();

  const int ri = idxs[i];
  const float4* grow =
      (const float4*)(g + ((size_t)b * NN + (size_t)ri) * NN);
  float4* rowv = (float4*)row;

#if TOPK_USE_ASYNC_LDS
  // Direct global -> LDS async copy (no VGPR staging); 512 x b128 = 8 KB.
  for (int j = t; j < NN / 4; j += 256) {
    __builtin_amdgcn_global_load_async_to_lds_b128(
        (__attribute__((address_space(1))) v4i*)(grow + j),
        (__attribute__((address_space(3))) v4i*)(rowv + j),
        /*offset=*/0, /*cpol=*/0);
  }
  __builtin_amdgcn_s_wait_asynccnt(0);
  __syncthreads();
#else
  for (int j = t; j < NN / 4; j += 256) rowv[j] = grow[j];
  __syncthreads();
#endif

  float* orow = gs_out + ((size_t)b * KK + (size_t)i) * KK;
  for (int j = t; j < KK; j += 256) orow[j] = row[idxs[j]];
}

// ---------------------------------------------------------------------------
extern "C" void kernel_launch(void* const* d_in, const int* in_sizes, int n_in,
                              void* d_out, int out_size, void* d_ws,
                              size_t ws_size, hipStream_t stream) {
  (void)in_sizes; (void)n_in; (void)out_size; (void)ws_size;
  const float* h      = (const float*)d_in[0];
  const float* g      = (const float*)d_in[1];
  const float* scores = (const float*)d_in[2];
  // d_in[3] is k (=512), fixed at compile time.

  float* out = (float*)d_out;
  float* hs = out;                                        // B*K*D
  float* gs = out + (size_t)BB * KK * DD;                 // B*K*K
  float* ss = out + (size_t)BB * KK * DD + (size_t)BB * KK * KK;  // B*K
  int* idx_ws = (int*)d_ws;                               // B*K ints (32 KB)

  topk_kernel<<<BB, 1024, 0, stream>>>(scores, idx_ws, ss);
  gather_h_kernel<<<BB * (KK / 16), 32, 0, stream>>>(h, idx_ws, hs);
  gather_g_kernel<<<dim3(KK, BB), 256, 0, stream>>>(g, idx_ws, gs);
}